// InterpolativeUpsampler_51453708206619
// MI455X (gfx1250) — compile-verified
//
#include <hip/hip_runtime.h>
#include <stdint.h>

// Problem constants (fixed by the reference): x is (4,256,128,128) f32,
// output is (4,256,256,256) f32, depthwise separable blur-upsample 2x.
#define IH 128
#define IW 128
#define OH 256
#define OW 256
#define TILE_O 64          // output tile edge
#define TILE_I 32          // input tile edge (TILE_O/2)
#define HALO   34          // TILE_I + 2 (one halo each side)
#define LDSW   36          // padded LDS row stride (floats)

// Native 2-wide float vector (HIP's float2 is a class; the nontemporal
// builtin requires a scalar or clang ext_vector type).
typedef float v2f __attribute__((ext_vector_type(2)));

#if defined(__has_builtin)
#  if __has_builtin(__builtin_amdgcn_s_wait_asynccnt)
#    define WAIT_ASYNC0() __builtin_amdgcn_s_wait_asynccnt(0)
#  else
#    define WAIT_ASYNC0() asm volatile("s_wait_asynccnt 0" ::: "memory")
#  endif
#else
#  define WAIT_ASYNC0() asm volatile("s_wait_asynccnt 0" ::: "memory")
#endif

__global__ __launch_bounds__(256) void
blur_upsample2x_kernel(const float* __restrict__ x, float* __restrict__ out)
{
    __shared__ float s[HALO * LDSW];

    const int tid   = threadIdx.x;
    const int tx    = blockIdx.x;   // tile col: 0..3
    const int ty    = blockIdx.y;   // tile row: 0..3
    const int plane = blockIdx.z;   // 0..N*C-1

    const float* xp = x   + (size_t)plane * (IH * IW);
    float*       op = out + (size_t)plane * (OH * OW);

    const int ix0 = tx * TILE_I;
    const int iy0 = ty * TILE_I;

    // ---------------------------------------------------------------
    // Stage 34x34 zero-padded input tile into LDS.
    // Valid elements: async global->LDS copy (ASYNCcnt path, no VGPR hop).
    // Out-of-image halo: plain LDS store of 0 (disjoint addresses, so the
    // DS-store / async-write streams never touch the same LDS word).
    // ---------------------------------------------------------------
    const uint64_t base = (uint64_t)(uintptr_t)xp;
#pragma unroll
    for (int i = 0; i < 5; ++i) {
        int e = tid + i * 256;
        if (e < HALO * HALO) {
            int ry = e / HALO;
            int rx = e - ry * HALO;
            int gy = iy0 - 1 + ry;
            int gx = ix0 - 1 + rx;
            bool valid = (gy >= 0) & (gy < IH) & (gx >= 0) & (gx < IW);
            if (valid) {
                uint32_t lds_addr = (uint32_t)(uintptr_t)&s[ry * LDSW + rx];
                uint32_t voff     = (uint32_t)((gy * IW + gx) * 4);
                // GVS mode: per-lane LDS dest VGPR, per-lane 32-bit byte
                // offset VGPR, uniform 64-bit SGPR base. EXEC-masked lanes
                // do not transfer.
                asm volatile("global_load_async_to_lds_b32 %0, %1, %2"
                             :: "v"(lds_addr), "v"(voff), "s"(base)
                             : "memory");
            } else {
                s[ry * LDSW + rx] = 0.0f;
            }
        }
    }
    WAIT_ASYNC0();        // all of this wave's async LDS writes have landed
    __syncthreads();      // (compiler adds DS waits for the zero-stores)

    // ---------------------------------------------------------------
    // Compute: each thread produces a 2x2 output block per step from a
    // 3x3 LDS neighborhood; separable vertical-then-horizontal filter.
    //   even: (a+b)/2      odd: (a + 6b + c)/8
    // Lanes 0..31 of a wave cover 32 consecutive input columns ->
    // two coalesced 256B 8-byte row stores per step.
    // ---------------------------------------------------------------
    const int lx = tid & 31;   // input col within tile: 0..31
    const int wy = tid >> 5;   // 0..7

#pragma unroll
    for (int p = 0; p < 4; ++p) {
        int py = wy + p * 8;   // input row within tile: 0..31

        const float* r0 = &s[(py    ) * LDSW + lx];
        const float* r1 = &s[(py + 1) * LDSW + lx];
        const float* r2 = &s[(py + 2) * LDSW + lx];

        float a0 = r0[0], a1 = r0[1], a2 = r0[2];
        float b0 = r1[0], b1 = r1[1], b2 = r1[2];
        float c0 = r2[0], c1 = r2[1], c2 = r2[2];

        // vertical pass (3 columns: m-1, m, m+1)
        float e0 = 0.5f   * (a0 + b0);
        float e1 = 0.5f   * (a1 + b1);
        float e2 = 0.5f   * (a2 + b2);
        float o0 = 0.125f * (a0 + 6.0f * b0 + c0);
        float o1 = 0.125f * (a1 + 6.0f * b1 + c1);
        float o2 = 0.125f * (a2 + 6.0f * b2 + c2);

        // horizontal pass -> 2x2 output block
        v2f top, bot;
        top.x = 0.5f   * (e0 + e1);
        top.y = 0.125f * (e0 + 6.0f * e1 + e2);
        bot.x = 0.5f   * (o0 + o1);
        bot.y = 0.125f * (o0 + 6.0f * o1 + o2);

        int oy = ty * TILE_O + 2 * py;
        int ox = tx * TILE_O + 2 * lx;

        // Output (256MB) exceeds L2 (192MB) and is never re-read:
        // stream it with non-temporal stores so the 9x-reused input
        // (64MB) stays resident in L2.
        __builtin_nontemporal_store(top, (v2f*)(op + (size_t)oy       * OW + ox));
        __builtin_nontemporal_store(bot, (v2f*)(op + (size_t)(oy + 1) * OW + ox));
    }
}

extern "C" void kernel_launch(void* const* d_in, const int* in_sizes, int n_in,
                              void* d_out, int out_size, void* d_ws, size_t ws_size,
                              hipStream_t stream) {
    (void)n_in; (void)out_size; (void)d_ws; (void)ws_size;
    const float* x   = (const float*)d_in[0];
    float*       out = (float*)d_out;
    const int planes = in_sizes[0] / (IH * IW);   // N*C = 1024
    dim3 grid(OW / TILE_O, OH / TILE_O, planes);  // (4, 4, 1024)
    hipLaunchKernelGGL(blur_upsample2x_kernel, grid, dim3(256), 0, stream, x, out);
}